// ConditionalAttentionMech_6803228197704
// MI455X (gfx1250) — compile-verified
//
#include <hip/hip_runtime.h>
#include <hip/hip_bf16.h>
#include <stdint.h>

#define N_ROWS  8192
#define IN_F    256
#define OUT_F   64
#define LEAK    0.2f
#define NEG_BIG (-9.0e15f)
#define SPLIT   4
#define JCHUNK  (N_ROWS / SPLIT)

typedef __attribute__((ext_vector_type(2))) float v2f;
typedef __attribute__((ext_vector_type(8))) float v8f;

// ---------------------------------------------------------------------------
// Kernel 1: pack adj (int32 0/1, 256MB) into a bitmask (8MB). This is the only
// kernel that touches the 256MB input; fully coalesced, wave32 ballot.
// ---------------------------------------------------------------------------
__global__ void k_pack_adj(const int* __restrict__ adj, uint32_t* __restrict__ mask) {
    int row  = blockIdx.x;
    int wave = threadIdx.x >> 5;
    int lane = threadIdx.x & 31;
    const int* ar = adj + (size_t)row * N_ROWS;
    uint32_t*  mr = mask + (size_t)row * (N_ROWS / 32);
    for (int w = wave; w < N_ROWS / 32; w += 8) {
        int v = ar[w * 32 + lane];
        uint32_t m = __builtin_amdgcn_ballot_w32(v > 0);
        if (lane == 0) mr[w] = m;
    }
}

// ---------------------------------------------------------------------------
// Kernel 2: h = x @ W  via V_WMMA_F32_16X16X4_F32.
// A (16x4 f32): lane = row (lane&15), K = v + 2*(lane>>4).
// B (4x16 f32): VGPR v, lanes 0-15 -> K=v, lanes 16-31 -> K=v+2, N = lane&15.
// C (16x16):    VGPR r, row = r + 8*(lane>>4), col = lane&15.
// One 16x16 output tile per wave; 512 row-tiles x 4 col-tiles.
// ---------------------------------------------------------------------------
__global__ __launch_bounds__(128) void k_gemm_h(const float* __restrict__ x,
                                                const float* __restrict__ W,
                                                float* __restrict__ h) {
    int lane = threadIdx.x & 31;
    int wave = threadIdx.x >> 5;
    int task = blockIdx.x * 4 + wave;         // 2048 tasks
    int rowTile = task >> 2;                  // 0..511
    int colTile = task & 3;                   // 0..3
    int i0 = rowTile * 16;
    int c0 = colTile * 16;
    int r    = lane & 15;
    int hgrp = lane >> 4;

    v8f acc = {};
    const float* xrow = x + (size_t)(i0 + r) * IN_F;
    for (int k = 0; k < IN_F; k += 4) {
        v2f a;
        a.x = xrow[k + 2 * hgrp];
        a.y = xrow[k + 2 * hgrp + 1];
        v2f b;
        b.x = W[(size_t)(k + 0 + 2 * hgrp) * OUT_F + c0 + r];
        b.y = W[(size_t)(k + 1 + 2 * hgrp) * OUT_F + c0 + r];
        acc = __builtin_amdgcn_wmma_f32_16x16x4_f32(false, a, false, b,
                                                    (short)0, acc, false, false);
    }
#pragma unroll
    for (int rr = 0; rr < 8; ++rr)
        h[(size_t)(i0 + rr + 8 * hgrp) * OUT_F + c0 + r] = acc[rr];
}

// ---------------------------------------------------------------------------
// Kernel 3: per-row projections s_i, s_j, gamma(+1), beta. One wave per row.
// ---------------------------------------------------------------------------
__global__ void k_proj(const float* __restrict__ h,
                       const float* __restrict__ a_i, const float* __restrict__ a_j,
                       const float* __restrict__ wg,  const float* __restrict__ wb,
                       float* __restrict__ s_i, float* __restrict__ s_j,
                       float* __restrict__ gam, float* __restrict__ bet) {
    int lane = threadIdx.x & 31;
    int wave = threadIdx.x >> 5;
    int row  = blockIdx.x * 8 + wave;
    const float* hr = h + (size_t)row * OUT_F;
    float h0 = hr[2 * lane], h1 = hr[2 * lane + 1];
    float d0 = h0 * a_i[2 * lane] + h1 * a_i[2 * lane + 1];
    float d1 = h0 * a_j[2 * lane] + h1 * a_j[2 * lane + 1];
    float d2 = h0 * wg[2 * lane] + h1 * wg[2 * lane + 1];
    float d3 = h0 * wb[2 * lane] + h1 * wb[2 * lane + 1];
#pragma unroll
    for (int off = 16; off; off >>= 1) {
        d0 += __shfl_xor(d0, off, 32);
        d1 += __shfl_xor(d1, off, 32);
        d2 += __shfl_xor(d2, off, 32);
        d3 += __shfl_xor(d3, off, 32);
    }
    if (lane == 0) { s_i[row] = d0; s_j[row] = d1; gam[row] = d2 + 1.0f; bet[row] = d3; }
}

// ---------------------------------------------------------------------------
// Kernel 4: fused masked flash-softmax + P@h.  One wave = 32 output rows
// (2 sub-tiles of 16) x one j-split of 2048 columns.  P is built directly in
// the WMMA A-fragment layout; accumulation via V_WMMA_F32_16X16X4_F32.
// ---------------------------------------------------------------------------
__device__ __forceinline__ void attn_subtile(float si, uint32_t mw, int half,
                                             const float sj[4][2], const v2f B[4][4],
                                             float& m, float& lsum, v8f acc[4],
                                             int lane) {
    int hgrp = lane >> 4;
    float e[4][2];
    float bm = NEG_BIG;
#pragma unroll
    for (int kc = 0; kc < 4; ++kc) {
#pragma unroll
        for (int v = 0; v < 2; ++v) {
            int c2 = 4 * kc + 2 * hgrp + v;              // column within 16-block
            float s  = si + sj[kc][v];
            float lr = s > 0.0f ? s : LEAK * s;          // leaky relu
            uint32_t bit = (mw >> (c2 + 16 * half)) & 1u;
            float ev = bit ? lr : NEG_BIG;               // keep reference semantics
            e[kc][v] = ev;
            bm = fmaxf(bm, ev);
        }
    }
    bm = fmaxf(bm, __shfl_xor(bm, 16, 32));              // full row max (keyed lane&15)
    float mnew = fmaxf(m, bm);
    if (__builtin_amdgcn_ballot_w32(mnew > m)) {         // wave-uniform rescale guard
        float sc = __expf(m - mnew);                     // per-row, keyed lane&15
#pragma unroll
        for (int rr = 0; rr < 8; ++rr) {
            float s0 = __shfl(sc, rr + 8 * hgrp, 32);    // scale for C-row rr+8*hgrp
#pragma unroll
            for (int ct = 0; ct < 4; ++ct) acc[ct][rr] *= s0;
        }
        lsum *= sc;
    }
    m = mnew;
    float ps = 0.0f;
    v2f pa[4];
#pragma unroll
    for (int kc = 0; kc < 4; ++kc) {
        float p0 = __expf(e[kc][0] - mnew);
        float p1 = __expf(e[kc][1] - mnew);
        pa[kc].x = p0; pa[kc].y = p1;
        ps += p0 + p1;
    }
    ps += __shfl_xor(ps, 16, 32);                        // row sum of P
    lsum += ps;
#pragma unroll
    for (int kc = 0; kc < 4; ++kc)
#pragma unroll
        for (int ct = 0; ct < 4; ++ct)
            acc[ct] = __builtin_amdgcn_wmma_f32_16x16x4_f32(false, pa[kc], false, B[kc][ct],
                                                            (short)0, acc[ct], false, false);
}

__global__ __launch_bounds__(128) void k_attn(const float* __restrict__ h,
                                              const uint32_t* __restrict__ mask,
                                              const float* __restrict__ s_i,
                                              const float* __restrict__ s_j,
                                              float* __restrict__ Opart,
                                              float* __restrict__ Mpart,
                                              float* __restrict__ Lpart) {
    int lane = threadIdx.x & 31;
    int wave = threadIdx.x >> 5;
    int task = blockIdx.x * 4 + wave;         // (N/32) * SPLIT tasks
    int rowTile = task >> 2;
    int split   = task & (SPLIT - 1);
    int i0 = rowTile * 32;
    int j_begin = split * JCHUNK;
    int j_end   = j_begin + JCHUNK;
    int r    = lane & 15;
    int hgrp = lane >> 4;

    float si0 = s_i[i0 + r];
    float si1 = s_i[i0 + 16 + r];
    float m0 = NEG_BIG, l0 = 0.0f, m1 = NEG_BIG, l1 = 0.0f;
    v8f acc0[4] = {{0}, {0}, {0}, {0}};
    v8f acc1[4] = {{0}, {0}, {0}, {0}};
    const uint32_t* mrow0 = mask + (size_t)(i0 + r) * (N_ROWS / 32);
    const uint32_t* mrow1 = mask + (size_t)(i0 + 16 + r) * (N_ROWS / 32);

    for (int jb = j_begin; jb < j_end; jb += 32) {
        uint32_t mw0 = mrow0[jb >> 5];
        uint32_t mw1 = mrow1[jb >> 5];
#pragma unroll
        for (int half = 0; half < 2; ++half) {
            int j16 = jb + 16 * half;
            v2f B[4][4];
            float sj[4][2];
#pragma unroll
            for (int kc = 0; kc < 4; ++kc) {
                int jr = j16 + 4 * kc + 2 * hgrp;        // B row for v=0 (v=1 -> jr+1)
                const float* hp = h + (size_t)jr * OUT_F;
#pragma unroll
                for (int ct = 0; ct < 4; ++ct) {
                    B[kc][ct].x = hp[ct * 16 + r];
                    B[kc][ct].y = hp[OUT_F + ct * 16 + r];
                }
                sj[kc][0] = s_j[jr];
                sj[kc][1] = s_j[jr + 1];
            }
            attn_subtile(si0, mw0, half, sj, B, m0, l0, acc0, lane);
            attn_subtile(si1, mw1, half, sj, B, m1, l1, acc1, lane);
        }
    }

#pragma unroll
    for (int st = 0; st < 2; ++st) {
        v8f*  acc = st ? acc1 : acc0;
        float mm  = st ? m1 : m0;
        float lv  = st ? l1 : l0;
#pragma unroll
        for (int ct = 0; ct < 4; ++ct)
#pragma unroll
            for (int rr = 0; rr < 8; ++rr) {
                int row = i0 + st * 16 + rr + 8 * hgrp;
                Opart[((size_t)row * SPLIT + split) * OUT_F + ct * 16 + r] = acc[ct][rr];
            }
        if (hgrp == 0) {
            int row = i0 + st * 16 + r;
            Mpart[row * SPLIT + split] = mm;
            Lpart[row * SPLIT + split] = lv;
        }
    }
}

// ---------------------------------------------------------------------------
// Kernel 5: merge the SPLIT partials (flash merge), normalize, apply FiLM.
// One wave per row; lane handles 2 columns.
// ---------------------------------------------------------------------------
__global__ void k_combine(const float* __restrict__ Opart, const float* __restrict__ Mpart,
                          const float* __restrict__ Lpart, const float* __restrict__ gam,
                          const float* __restrict__ bet, float* __restrict__ out) {
    int lane = threadIdx.x & 31;
    int wave = threadIdx.x >> 5;
    int row  = blockIdx.x * 8 + wave;
    float M[SPLIT], L[SPLIT];
    float ms = NEG_BIG;
#pragma unroll
    for (int s = 0; s < SPLIT; ++s) {
        M[s] = Mpart[row * SPLIT + s];
        L[s] = Lpart[row * SPLIT + s];
        ms = fmaxf(ms, M[s]);
    }
    float w[SPLIT], lsum = 0.0f;
#pragma unroll
    for (int s = 0; s < SPLIT; ++s) { w[s] = __expf(M[s] - ms); lsum += w[s] * L[s]; }
    float o0 = 0.0f, o1 = 0.0f;
    int c = 2 * lane;
#pragma unroll
    for (int s = 0; s < SPLIT; ++s) {
        const float* op = Opart + ((size_t)row * SPLIT + s) * OUT_F;
        o0 += w[s] * op[c];
        o1 += w[s] * op[c + 1];
    }
    float inv = 1.0f / lsum;
    float g = gam[row], b = bet[row];
    out[(size_t)row * OUT_F + c]     = g * o0 * inv + b;
    out[(size_t)row * OUT_F + c + 1] = g * o1 * inv + b;
}

// ---------------------------------------------------------------------------
extern "C" void kernel_launch(void* const* d_in, const int* in_sizes, int n_in,
                              void* d_out, int out_size, void* d_ws, size_t ws_size,
                              hipStream_t stream) {
    (void)in_sizes; (void)n_in; (void)out_size; (void)ws_size;
    const float* x   = (const float*)d_in[0];
    const int*   adj = (const int*)d_in[1];
    const float* W   = (const float*)d_in[2];
    const float* a_i = (const float*)d_in[3];
    const float* a_j = (const float*)d_in[4];
    const float* w_g = (const float*)d_in[5];
    const float* w_b = (const float*)d_in[6];
    float* out = (float*)d_out;

    char* ws = (char*)d_ws;
    size_t off = 0;
    float* h    = (float*)(ws + off); off += (size_t)N_ROWS * OUT_F * 4;     // 2 MB
    float* s_i  = (float*)(ws + off); off += (size_t)N_ROWS * 4;
    float* s_j  = (float*)(ws + off); off += (size_t)N_ROWS * 4;
    float* gam  = (float*)(ws + off); off += (size_t)N_ROWS * 4;
    float* bet  = (float*)(ws + off); off += (size_t)N_ROWS * 4;
    uint32_t* mask = (uint32_t*)(ws + off); off += (size_t)N_ROWS * (N_ROWS / 32) * 4; // 8 MB
    float* Opart = (float*)(ws + off); off += (size_t)N_ROWS * SPLIT * OUT_F * 4;      // 8 MB
    float* Mpart = (float*)(ws + off); off += (size_t)N_ROWS * SPLIT * 4;
    float* Lpart = (float*)(ws + off); off += (size_t)N_ROWS * SPLIT * 4;

    k_pack_adj<<<N_ROWS, 256, 0, stream>>>(adj, mask);
    k_gemm_h<<<(N_ROWS / 16) * (OUT_F / 16) / 4, 128, 0, stream>>>(x, W, h);
    k_proj<<<N_ROWS / 8, 256, 0, stream>>>(h, a_i, a_j, w_g, w_b, s_i, s_j, gam, bet);
    k_attn<<<(N_ROWS / 32) * SPLIT / 4, 128, 0, stream>>>(h, mask, s_i, s_j, Opart, Mpart, Lpart);
    k_combine<<<N_ROWS / 8, 256, 0, stream>>>(Opart, Mpart, Lpart, gam, bet, out);
}